// LSTMLM_72421738545579
// MI455X (gfx1250) — compile-verified
//
#include <hip/hip_runtime.h>
#include <math.h>

#define Vv 50257
#define Ee 128
#define Hh 20
#define Ll 5
#define Nt 256
#define Gg 80            // 4*H gates
#define Kk (Nt * Hh)     // 5120 = flat length

typedef float v2f __attribute__((ext_vector_type(2)));
typedef float v8f __attribute__((ext_vector_type(8)));

__device__ __forceinline__ float sigm(float x) { return 1.0f / (1.0f + expf(-x)); }

// ---------------------------------------------------------------------------
// Kernel 1: layer-pipelined LSTM (1 workgroup, 512 threads = 16 waves).
// Wavefront step s: layer l handles timestep t = s - l. Critical path = 260
// steps instead of 1280. Rest-layer weights in LDS; w_ih0 (40KB) from cache.
// ---------------------------------------------------------------------------
__global__ __launch_bounds__(512)
void lstm_pipeline_kernel(const int* __restrict__ batch,
                          const float* __restrict__ h0, const float* __restrict__ c0,
                          const float* __restrict__ emb,
                          const float* __restrict__ w_ih0, const float* __restrict__ w_hh0,
                          const float* __restrict__ b_ih0, const float* __restrict__ b_hh0,
                          const float* __restrict__ w_ih_r, const float* __restrict__ w_hh_r,
                          const float* __restrict__ b_ih_r, const float* __restrict__ b_hh_r,
                          float* __restrict__ hn_cn_out, float* __restrict__ flat)
{
    __shared__ float s_whh0[Gg * Hh];          //  6.4 KB
    __shared__ float s_wih_r[4 * Gg * Hh];     // 25.6 KB
    __shared__ float s_whh_r[4 * Gg * Hh];     // 25.6 KB
    __shared__ float s_bias[Ll * Gg];          // b_ih + b_hh pre-summed
    __shared__ float s_xemb[2][Ee];            // double-buffered embedding row
    __shared__ float s_xbuf[Ll][2][Hh];        // per-layer output double buffer
    __shared__ float s_gbuf[Ll][Gg];           // gate pre-activations
    __shared__ float s_h[Ll][Hh];
    __shared__ float s_c[Ll][Hh];

    const int tid = threadIdx.x;

    // cooperative weight staging (~60 KB into LDS)
    for (int i = tid; i < Gg * Hh; i += 512) s_whh0[i] = w_hh0[i];
    for (int i = tid; i < 4 * Gg * Hh; i += 512) { s_wih_r[i] = w_ih_r[i]; s_whh_r[i] = w_hh_r[i]; }
    for (int i = tid; i < Gg; i += 512)     s_bias[i]      = b_ih0[i]  + b_hh0[i];
    for (int i = tid; i < 4 * Gg; i += 512) s_bias[Gg + i] = b_ih_r[i] + b_hh_r[i];
    for (int i = tid; i < Ll * Hh; i += 512) { (&s_h[0][0])[i] = h0[i]; (&s_c[0][0])[i] = c0[i]; }
    if (tid < Ee) s_xemb[0][tid] = emb[(size_t)batch[0] * Ee + tid];
    __syncthreads();

    for (int s = 0; s < Nt + Ll - 1; ++s) {
        // ---- phase 1: gate pre-activations (+ embedding prefetch) ----
        if (tid < 2 * Gg) {
            // layer 0: 2 threads per gate split the E=128 input dot
            const int t = s;
            if (t < Nt) {
                const int g = tid >> 1, half = tid & 1;
                const float4* w4 = (const float4*)(w_ih0 + g * Ee + half * 64);
                const float4* x4 = (const float4*)(&s_xemb[t & 1][half * 64]);
                float a0 = 0.f, a1 = 0.f, a2 = 0.f, a3 = 0.f;
#pragma unroll
                for (int q = 0; q < 16; ++q) {
                    float4 w = w4[q], x = x4[q];
                    a0 = fmaf(w.x, x.x, a0); a1 = fmaf(w.y, x.y, a1);
                    a2 = fmaf(w.z, x.z, a2); a3 = fmaf(w.w, x.w, a3);
                }
                float sum = (a0 + a2) + (a1 + a3);
                sum += __shfl_xor(sum, 1, 32);   // partner in same wave32
                if (half == 0) {
                    const float4* wh = (const float4*)(&s_whh0[g * Hh]);
                    const float4* hv = (const float4*)(&s_h[0][0]);
                    float b0 = 0.f, b1 = 0.f;
#pragma unroll
                    for (int q = 0; q < 5; ++q) {
                        float4 w = wh[q], h = hv[q];
                        b0 = fmaf(w.x, h.x, b0); b1 = fmaf(w.y, h.y, b1);
                        b0 = fmaf(w.z, h.z, b0); b1 = fmaf(w.w, h.w, b1);
                    }
                    s_gbuf[0][g] = sum + b0 + b1 + s_bias[g];
                }
            }
        } else if (tid < 6 * Gg) {
            // layers 1..4: 1 thread per gate (K = 20 + 20)
            const int l = 1 + (tid - 2 * Gg) / Gg;
            const int g = (tid - 2 * Gg) % Gg;
            const int t = s - l;
            if (t >= 0 && t < Nt) {
                const float4* wi = (const float4*)(&s_wih_r[((l - 1) * Gg + g) * Hh]);
                const float4* wh = (const float4*)(&s_whh_r[((l - 1) * Gg + g) * Hh]);
                const float4* xv = (const float4*)(&s_xbuf[l - 1][t & 1][0]);
                const float4* hv = (const float4*)(&s_h[l][0]);
                float b0 = 0.f, b1 = 0.f;
#pragma unroll
                for (int q = 0; q < 5; ++q) {
                    float4 a = wi[q], x = xv[q], b = wh[q], h = hv[q];
                    b0 = fmaf(a.x, x.x, b0); b1 = fmaf(b.x, h.x, b1);
                    b0 = fmaf(a.y, x.y, b0); b1 = fmaf(b.y, h.y, b1);
                    b0 = fmaf(a.z, x.z, b0); b1 = fmaf(b.z, h.z, b1);
                    b0 = fmaf(a.w, x.w, b0); b1 = fmaf(b.w, h.w, b1);
                }
                s_gbuf[l][g] = b0 + b1 + s_bias[l * Gg + g];
            }
        } else {
            // prefetch wave (tid 480..511): stage next embedding row
            const int tn = s + 1;
            if (tn < Nt) {
                const int lane = tid - 480;
                const size_t rowbase = (size_t)batch[tn] * Ee;
                for (int e = lane; e < Ee; e += 32) s_xemb[tn & 1][e] = emb[rowbase + e];
            }
        }
        __syncthreads();

        // ---- phase 2: elementwise LSTM cell update (100 threads) ----
        if (tid < Ll * Hh) {
            const int l = tid / Hh, j = tid % Hh;
            const int t = s - l;
            if (t >= 0 && t < Nt) {
                const float gi = s_gbuf[l][j];
                const float gf = s_gbuf[l][Hh + j];
                const float gg = s_gbuf[l][2 * Hh + j];
                const float go = s_gbuf[l][3 * Hh + j];
                float cc = sigm(gf) * s_c[l][j] + sigm(gi) * tanhf(gg);
                float hh = sigm(go) * tanhf(cc);
                s_c[l][j] = cc; s_h[l][j] = hh;
                s_xbuf[l][t & 1][j] = hh;
                if (l == Ll - 1) flat[t * Hh + j] = hh;             // feed for kernel 2
                if (t == Nt - 1) {                                   // hn / cn outputs
                    hn_cn_out[l * Hh + j] = hh;
                    hn_cn_out[Ll * Hh + l * Hh + j] = cc;
                }
            }
        }
        __syncthreads();
    }
}

// ---------------------------------------------------------------------------
// Kernel 2: logits = lin_w @ flat + lin_b via V_WMMA_F32_16X16X4_F32.
// 1.03 GB streamed once -> ~44us floor @ 23.3 TB/s. Each wave owns 16 rows.
// A tile (16x4 fp32): lane = M (lane&15), K-pair chosen by lane>=16.
// B (4x16) broadcasts flat across all N columns (memory-bound; waste is free).
// lin_w read with non-temporal loads (no cache pollution); flat staged in LDS.
// ---------------------------------------------------------------------------
__global__ __launch_bounds__(256)
void logits_wmma_kernel(const float* __restrict__ lin_w, const float* __restrict__ lin_b,
                        const float* __restrict__ flat, float* __restrict__ logits)
{
    __shared__ float s_flat[Kk];                 // 20 KB
    for (int i = threadIdx.x; i < Kk; i += 256) s_flat[i] = flat[i];
    __syncthreads();

    const int lane  = threadIdx.x & 31;
    const int wave  = threadIdx.x >> 5;          // 8 waves per block
    const int tile  = blockIdx.x * 8 + wave;     // 16-row tile
    const int m     = lane & 15;                 // A row for this lane
    const int khalf = lane >> 4;                 // 0: K={0,1}, 1: K={2,3}

    long long row  = (long long)tile * 16 + m;
    long long rowc = (row < Vv) ? row : (Vv - 1);
    const float* wrow = lin_w + rowc * (long long)Kk + 2 * khalf;

    v8f acc = {0.f, 0.f, 0.f, 0.f, 0.f, 0.f, 0.f, 0.f};
    for (int k = 0; k < Kk; k += 16) {
#pragma unroll
        for (int u = 0; u < 4; ++u) {
            const int kk = k + 4 * u;
            v2f a = __builtin_nontemporal_load((const v2f*)(wrow + kk));   // 8B/lane stream
            v2f b;
            b.x = s_flat[kk + 2 * khalf];        // B VGPR0: K = 0 + 2*half
            b.y = s_flat[kk + 2 * khalf + 1];    // B VGPR1: K = 1 + 2*half
            acc = __builtin_amdgcn_wmma_f32_16x16x4_f32(
                false, a, false, b, (short)0, acc, false, false);
        }
    }

    // D layout: lane 0 holds M=0..7 in acc[0..7]; lane 16 holds M=8..15.
    if ((lane & 15) == 0) {
        const long long base = (long long)tile * 16 + khalf * 8;
#pragma unroll
        for (int r = 0; r < 8; ++r) {
            const long long rr = base + r;
            if (rr < Vv) logits[rr] = acc[r] + lin_b[rr];
        }
    }
}

// ---------------------------------------------------------------------------
extern "C" void kernel_launch(void* const* d_in, const int* in_sizes, int n_in,
                              void* d_out, int out_size, void* d_ws, size_t ws_size,
                              hipStream_t stream) {
    const int*   batch  = (const int*)  d_in[0];
    const float* h0     = (const float*)d_in[1];
    const float* c0     = (const float*)d_in[2];
    const float* emb    = (const float*)d_in[3];
    const float* w_ih0  = (const float*)d_in[4];
    const float* w_hh0  = (const float*)d_in[5];
    const float* b_ih0  = (const float*)d_in[6];
    const float* b_hh0  = (const float*)d_in[7];
    const float* w_ih_r = (const float*)d_in[8];
    const float* w_hh_r = (const float*)d_in[9];
    const float* b_ih_r = (const float*)d_in[10];
    const float* b_hh_r = (const float*)d_in[11];
    const float* lin_w  = (const float*)d_in[12];
    const float* lin_b  = (const float*)d_in[13];

    float* out  = (float*)d_out;        // [hn(100) | cn(100) | logits(50257)]
    float* flat = (float*)d_ws;         // 5120 floats scratch

    lstm_pipeline_kernel<<<1, 512, 0, stream>>>(
        batch, h0, c0, emb, w_ih0, w_hh0, b_ih0, b_hh0,
        w_ih_r, w_hh_r, b_ih_r, b_hh_r, out, flat);

    const int tiles  = (Vv + 15) / 16;       // 3142
    const int blocks = (tiles + 7) / 8;      // 393
    logits_wmma_kernel<<<blocks, 256, 0, stream>>>(lin_w, lin_b, flat, out + 2 * Ll * Hh);
}